// ChamferDistance_65266323030088
// MI455X (gfx1250) — compile-verified
//
#include <hip/hip_runtime.h>
#include <float.h>

typedef __attribute__((ext_vector_type(2))) float v2f;
typedef __attribute__((ext_vector_type(8))) float v8f;

#define BATCH 4
#define NPTS 8192
#define TILE 16
#define WAVES_PER_BLOCK 8
#define THREADS (WAVES_PER_BLOCK * 32)

// For each 16-row tile of X, compute min_j ||x_i - y_j||^2 over ALL columns j,
// using V_WMMA_F32_16X16X4_F32 (fp32 matrix pipe, K=4 padded from D=3).
//
// A[M,k] = (-2x0, -2x1, -2x2, ||x_M||^2)  (norm folded into the K=3 pad slot)
// B[k,N] = ( y0,   y1,   y2,  1 )
// => (A*B)[M,N] = ||x_M||^2 - 2 x_M.y_N ;  dist = tile + ||y_N||^2
__global__ __launch_bounds__(THREADS) void chamfer_rowmin_wmma(
    const float* __restrict__ X, const float* __restrict__ Y,
    float* __restrict__ rowMin)
{
    const int wave = threadIdx.x >> 5;
    const int lane = threadIdx.x & 31;
    const int half = lane >> 4;   // 0: carries K=0,1   1: carries K=2,3
    const int m    = lane & 15;   // row (A) / column (B) index within tile

    const int tilesPerBatch = NPTS / TILE;                 // 512
    const int tile = blockIdx.x * WAVES_PER_BLOCK + wave;  // global row-tile id
    const int b    = tile / tilesPerBatch;
    const int r0   = (tile % tilesPerBatch) * TILE;

    const float* __restrict__ Xb = X + (size_t)b * NPTS * 3;
    const float* __restrict__ Yb = Y + (size_t)b * NPTS * 3;

    // ---- Build A fragment (per-lane v2f), 16x4 f32 layout:
    // lanes 0-15: (A[m][0], A[m][1]) ; lanes 16-31: (A[m][2], A[m][3])
    const float x0  = Xb[(r0 + m) * 3 + 0];
    const float x1  = Xb[(r0 + m) * 3 + 1];
    const float x2  = Xb[(r0 + m) * 3 + 2];
    const float xsq = x0 * x0 + x1 * x1 + x2 * x2;

    v2f a;
    a.x = half ? (-2.0f * x2) : (-2.0f * x0);
    a.y = half ? xsq          : (-2.0f * x1);

    float runmin[8];
#pragma unroll
    for (int v = 0; v < 8; ++v) runmin[v] = FLT_MAX;

    for (int jt = 0; jt < tilesPerBatch; ++jt) {
        const int col = jt * TILE + m;
        // Load the full y column (12 contiguous bytes) so the column norm is
        // computed locally -- keeps LDS/bpermute out of the hot loop.
        const float y0 = Yb[col * 3 + 0];
        const float y1 = Yb[col * 3 + 1];
        const float y2 = Yb[col * 3 + 2];
        const float ysq = y0 * y0 + y1 * y1 + y2 * y2;

        // B fragment (4x16 f32): lanes 0-15 hold (B[0][m], B[1][m]) = (y0,y1),
        // lanes 16-31 hold (B[2][m], B[3][m]) = (y2, 1)
        v2f bm;
        bm.x = half ? y2   : y0;
        bm.y = half ? 1.0f : y1;

        v8f c = {};
        // (neg_a, A, neg_b, B, c_mod, C, reuse_a, reuse_b)
        c = __builtin_amdgcn_wmma_f32_16x16x4_f32(
                false, a, false, bm, (short)0, c, false, false);

        // D tile layout: c[v] -> row M = v + 8*half, col N = m
#pragma unroll
        for (int v = 0; v < 8; ++v) {
            runmin[v] = fminf(runmin[v], c[v] + ysq);
        }
    }

    // min over the 16 columns held across lanes of each half-wave
#pragma unroll
    for (int v = 0; v < 8; ++v) {
        float r = runmin[v];
        r = fminf(r, __shfl_xor(r, 8, 32));
        r = fminf(r, __shfl_xor(r, 4, 32));
        r = fminf(r, __shfl_xor(r, 2, 32));
        r = fminf(r, __shfl_xor(r, 1, 32));
        runmin[v] = r;
    }

    if (m == 0) {  // lane 0 -> rows 0..7, lane 16 -> rows 8..15
        float* out = rowMin + (size_t)b * NPTS + r0 + half * 8;
#pragma unroll
        for (int v = 0; v < 8; ++v) out[v] = runmin[v];
    }
}

// Single-block sum of minX ++ minY, then scale: mean(minX)+mean(minY)
__global__ __launch_bounds__(256) void chamfer_mean_kernel(
    const float* __restrict__ mins, float* __restrict__ out, int total)
{
    __shared__ float sred[256];
    float s = 0.0f;
    for (int i = threadIdx.x; i < total; i += 256) s += mins[i];
    sred[threadIdx.x] = s;
    __syncthreads();
    for (int off = 128; off > 0; off >>= 1) {
        if (threadIdx.x < off) sred[threadIdx.x] += sred[threadIdx.x + off];
        __syncthreads();
    }
    if (threadIdx.x == 0) out[0] = sred[0] / (float)(BATCH * NPTS);
}

extern "C" void kernel_launch(void* const* d_in, const int* in_sizes, int n_in,
                              void* d_out, int out_size, void* d_ws, size_t ws_size,
                              hipStream_t stream) {
    const float* x = (const float*)d_in[0];
    const float* y = (const float*)d_in[1];
    float* minX = (float*)d_ws;                  // BATCH*NPTS floats
    float* minY = minX + (size_t)BATCH * NPTS;   // BATCH*NPTS floats

    dim3 grid((BATCH * (NPTS / TILE)) / WAVES_PER_BLOCK);  // 256 blocks
    chamfer_rowmin_wmma<<<grid, THREADS, 0, stream>>>(x, y, minX);
    chamfer_rowmin_wmma<<<grid, THREADS, 0, stream>>>(y, x, minY);
    chamfer_mean_kernel<<<1, 256, 0, stream>>>(minX, (float*)d_out,
                                               2 * BATCH * NPTS);
}